// KNN_dist_91225105367770
// MI455X (gfx1250) — compile-verified
//
#include <hip/hip_runtime.h>
#include <math.h>

typedef __attribute__((ext_vector_type(2))) float v2f;
typedef __attribute__((ext_vector_type(8))) float v8f;

#define KSEL   20
#define NPTS   8192
#define NTILES (NPTS / 16)     // 512 candidate tiles of 16
#define WAVES  4               // waves per workgroup, each scans NTILES/WAVES tiles
#define TPW    (NTILES / WAVES)

struct P3 { float x, y, z; };

// Branchless sorted insert (ascending by (key, arrival order)) under a 1-compare guard.
// Key shift uses bd[j] = max(min(d, bd[j]), bd[j-1]) (lowers to v_minmax_num_f32);
// ties keep the earlier (smaller) index via strict '<' everywhere, matching
// lax.top_k tie-breaking since candidates arrive in increasing index per lane.
__device__ __forceinline__ void insert20(float (&bd)[KSEL], int (&bi)[KSEL],
                                         float d, int i) {
    if (d < bd[KSEL - 1]) {
#pragma unroll
        for (int j = KSEL - 1; j >= 1; --j) {
            const bool up  = d < bd[j - 1];          // shift level j
            const bool dlt = d < bd[j];              // insert at level j
            const int  ni  = up ? bi[j - 1] : i;
            bi[j] = (up || dlt) ? ni : bi[j];
            bd[j] = fmaxf(fminf(d, bd[j]), bd[j - 1]);
        }
        const bool c0 = d < bd[0];
        bi[0] = c0 ? i : bi[0];
        bd[0] = fminf(d, bd[0]);
    }
}

// A (16x4, candidates): row n = (x, y, z, |x|^2)
// B (4x16, queries):    col q = (-2qx, -2qy, -2qz, 1)
// D[n,q] = |x_n|^2 - 2 x_n.q  ==  dist^2 minus the per-query constant |q|^2
// (constant per column -> identical top-k ordering).
__global__ __launch_bounds__(32 * WAVES)
void knn_wmma_kernel(const float* __restrict__ verts, int* __restrict__ out) {
    const int  tid   = threadIdx.x;
    const int  wave  = tid >> 5;
    const int  lane  = tid & 31;
    const int  col   = lane & 15;
    const bool hi    = lane >= 16;
    const int  qbase = blockIdx.x * 16;

    // Loop-invariant B operand for this block's 16 queries (same in every wave).
    const int   q  = qbase + col;
    const float qx = verts[q * 3 + 0];
    const float qy = verts[q * 3 + 1];
    const float qz = verts[q * 3 + 2];
    v2f b;
    b.x = hi ? (-2.0f * qz) : (-2.0f * qx);   // K2 : K0
    b.y = hi ? 1.0f         : (-2.0f * qy);   // K3 : K1

    float bd[KSEL];
    int   bi[KSEL];
#pragma unroll
    for (int j = 0; j < KSEL; ++j) { bd[j] = INFINITY; bi[j] = 0x7fffffff; }

    const int mofs = hi ? 8 : 0;   // C layout: VGPR v -> row v (lanes 0-15) / v+8 (16-31)
    const int t0   = wave * TPW;
    const int tend = t0 + TPW;

    auto loadP = [&](int t) {
        const int n = t * 16 + col;
        P3 p;
        p.x = verts[n * 3 + 0];
        p.y = verts[n * 3 + 1];
        p.z = verts[n * 3 + 2];
        return p;
    };

    // Process one candidate tile: WMMA produces 8 keys per lane for this lane's query;
    // a 3-level min-tree rejects the whole tile in ~9 issues in the steady state.
    auto processTile = [&](const P3& p, int t) {
        const float nn = fmaf(p.x, p.x, fmaf(p.y, p.y, p.z * p.z));
        v2f a;
        a.x = hi ? p.z : p.x;
        a.y = hi ? nn  : p.y;
        v8f c = {};
        c = __builtin_amdgcn_wmma_f32_16x16x4_f32(false, a, false, b,
                                                  (short)0, c, false, false);
        const float m01 = fminf(c[0], c[1]);
        const float m23 = fminf(c[2], c[3]);
        const float m45 = fminf(c[4], c[5]);
        const float m67 = fminf(c[6], c[7]);
        const float tmin = fminf(fminf(m01, m23), fminf(m45, m67));
        if (tmin < bd[KSEL - 1]) {
            const int base = t * 16 + mofs;
#pragma unroll
            for (int v = 0; v < 8; ++v) insert20(bd, bi, c[v], base + v);
        }
    };

    // Software pipeline: loads issued two tiles ahead of use.
    P3 p0 = loadP(t0);
    P3 p1 = loadP(t0 + 1);

    for (int t = t0; t < tend; t += 2) {
        const int tn0 = (t + 2 < tend) ? (t + 2) : t0;   // wrapped tail loads (unused)
        const int tn1 = (t + 3 < tend) ? (t + 3) : t0;
        P3 n0 = loadP(tn0);
        P3 n1 = loadP(tn1);
        // Locality 3 -> WGP-scope prefetch: pull tile t+4 into the near caches.
        __builtin_prefetch(&verts[((t + 4 < tend) ? (t + 4) : t0) * 48 + col * 3], 0, 3);

        processTile(p0, t);
        processTile(p1, t + 1);

        p0 = n0;
        p1 = n1;
    }

    // Each query now has 8 sorted 20-lists: (wave 0..3) x (lane, lane+16).
    // Dump all lists to LDS and 8-way merge with 16 threads.
    __shared__ float s_d[32 * WAVES][KSEL];
    __shared__ int   s_i[32 * WAVES][KSEL];
#pragma unroll
    for (int j = 0; j < KSEL; ++j) { s_d[tid][j] = bd[j]; s_i[tid][j] = bi[j]; }
    __syncthreads();

    if (tid < 16) {
        int p[8];
#pragma unroll
        for (int j = 0; j < 8; ++j) p[j] = 0;
        const int qg = qbase + tid;
        for (int k = 0; k < KSEL; ++k) {
            float bestd = INFINITY;
            int   besti = 0x7fffffff;
            int   bl    = 0;
#pragma unroll
            for (int j = 0; j < 8; ++j) {
                const int row = (j >> 1) * 32 + tid + ((j & 1) << 4);
                const int pj  = p[j];
                const float dj = (pj < KSEL) ? s_d[row][pj] : INFINITY;
                const int   ij = (pj < KSEL) ? s_i[row][pj] : 0x7fffffff;
                const bool better = (dj < bestd) || (dj == bestd && ij < besti);
                bestd = better ? dj : bestd;
                besti = better ? ij : besti;
                bl    = better ? j  : bl;
            }
            out[qg * KSEL + k] = besti;
#pragma unroll
            for (int j = 0; j < 8; ++j) p[j] += (j == bl) ? 1 : 0;
        }
    }
}

extern "C" void kernel_launch(void* const* d_in, const int* in_sizes, int n_in,
                              void* d_out, int out_size, void* d_ws, size_t ws_size,
                              hipStream_t stream) {
    (void)in_sizes; (void)n_in; (void)d_ws; (void)ws_size; (void)out_size;
    // d_in[0] = feats [4,8192,64] (dead in the reference's live path)
    // d_in[1] = vertices [4,8192,3]; only batch 0 reaches the output.
    const float* verts = (const float*)d_in[1];
    int* out = (int*)d_out;   // [8192, 20] int32 indices

    knn_wmma_kernel<<<NPTS / 16, 32 * WAVES, 0, stream>>>(verts, out);
}